// BBoxEncoder_25821343383806
// MI455X (gfx1250) — compile-verified
//
#include <hip/hip_runtime.h>

typedef float v2f __attribute__((ext_vector_type(2)));
typedef float v8f __attribute__((ext_vector_type(8)));

#define ENC_DIM   32
#define N_POINTS  16
#define N_LEVELS  8
#define HIST_W    64
#define LVL_FEAT  (N_POINTS * ENC_DIM)            // 512
#define RAY_FEAT  (N_LEVELS * LVL_FEAT)           // 4096

// ---------------------------------------------------------------------------
// Kernel 1: max over depth[] via wave reduction + atomicMax into workspace.
// ---------------------------------------------------------------------------
__global__ void __launch_bounds__(256)
depth_max_kernel(const int* __restrict__ depth, int n, int* __restrict__ out) {
    int i = blockIdx.x * blockDim.x + threadIdx.x;
    int v = (i < n) ? depth[i] : 0;
#pragma unroll
    for (int off = 16; off > 0; off >>= 1)
        v = max(v, __shfl_down(v, off, 32));
    if ((threadIdx.x & 31) == 0) atomicMax(out, v);
}

// ---------------------------------------------------------------------------
// Kernel 2: one wave32 per (ray, level) tile.
//   A = corner weights  (16 points x 8 corners), f32
//   B = bbox embedding  (8 corners x 32 dims),   f32
//   D = A x B           (16 x 32) via 4x V_WMMA_F32_16X16X4_F32
// Block = 256 threads = 8 waves = the 8 levels of one ray; grid = n_rays.
// No divergent control flow -> EXEC all ones for WMMA.
// ---------------------------------------------------------------------------
__global__ void __launch_bounds__(256)
bbox_encode_kernel(const float* __restrict__ inp,
                   const int*   __restrict__ history,
                   const float* __restrict__ bbox_emb,
                   const float* __restrict__ nodes_min,
                   const float* __restrict__ nodes_max,
                   const int*   __restrict__ d_maxdepth,
                   float*       __restrict__ out)
{
    const int lane  = threadIdx.x & 31;
    const int wave  = threadIdx.x >> 5;
    const int ray   = blockIdx.x;           // tile>>3 == blockIdx.x
    const int level = wave;                 // tile&7  == wave

    int maxd = *d_maxdepth;                 // uniform -> scalar load
    maxd = maxd > N_LEVELS ? N_LEVELS : maxd;
    const float scale = (level < maxd) ? 1.0f : 0.0f;

    const int p    = lane & 15;             // point index (matrix row M base)
    const int half = lane >> 4;             // lane half selects K pair / M+8
    const int kb   = half * 2;              // K base within a 4-wide K group

    const int idx = history[ray * HIST_W + level];

    // node bbox (broadcast loads, L2/L0 absorb the redundancy)
    const float mn0 = nodes_min[idx * 3 + 0];
    const float mn1 = nodes_min[idx * 3 + 1];
    const float mn2 = nodes_min[idx * 3 + 2];
    float e0 = nodes_max[idx * 3 + 0] - mn0;
    float e1 = nodes_max[idx * 3 + 1] - mn1;
    float e2 = nodes_max[idx * 3 + 2] - mn2;
    e0 = (e0 == 0.0f) ? 0.5f : e0;
    e1 = (e1 == 0.0f) ? 0.5f : e1;
    e2 = (e2 == 0.0f) ? 0.5f : e2;

    // this lane's point coordinates (both lane-halves use the same point)
    const float* ip = inp + ((size_t)ray * N_POINTS + p) * 3;
    float xd = fminf(fmaxf((ip[0] - mn0) / e0, 0.0f), 1.0f);
    float yd = fminf(fmaxf((ip[1] - mn1) / e1, 0.0f), 1.0f);
    float zd = fminf(fmaxf((ip[2] - mn2) / e2, 0.0f), 1.0f);
    const float ux = 1.0f - xd, uy = 1.0f - yd, uz = 1.0f - zd;

    // corner order: 000,100,010,001,101,011,110,111
    const float w0 = ux * uy * uz;
    const float w1 = xd * uy * uz;
    const float w2 = ux * yd * uz;
    const float w3 = ux * uy * zd;
    const float w4 = xd * uy * zd;
    const float w5 = ux * yd * zd;
    const float w6 = xd * yd * uz;
    const float w7 = xd * yd * zd;

    // A-matrix (16x4 f32): lanes 0-15 -> {K0,K1}, lanes 16-31 -> {K2,K3}
    v2f a0, a1;
    a0.x = half ? w2 : w0;  a0.y = half ? w3 : w1;   // corners 0..3
    a1.x = half ? w6 : w4;  a1.y = half ? w7 : w5;   // corners 4..7

    const float* f = bbox_emb + (size_t)idx * (8 * ENC_DIM);
    const int col = lane & 15;                        // B/D column N
    float* orow = out + (size_t)ray * RAY_FEAT + (size_t)level * LVL_FEAT;

#pragma unroll
    for (int nh = 0; nh < 2; ++nh) {                  // two 16-wide N halves
        const int c = nh * 16 + col;
        // B-matrix (4x16 f32): VGPR0 row K=kb, VGPR1 row K=kb+1
        v2f b0, b1;
        b0.x = f[(0 + kb) * ENC_DIM + c];
        b0.y = f[(1 + kb) * ENC_DIM + c];
        b1.x = f[(4 + kb) * ENC_DIM + c];
        b1.y = f[(5 + kb) * ENC_DIM + c];

        v8f acc = {0.f, 0.f, 0.f, 0.f, 0.f, 0.f, 0.f, 0.f};
        acc = __builtin_amdgcn_wmma_f32_16x16x4_f32(
                  false, a0, false, b0, (short)0, acc, false, false);
        acc = __builtin_amdgcn_wmma_f32_16x16x4_f32(
                  false, a1, false, b1, (short)0, acc, false, false);

        // D layout: VGPR v, lanes 0-15 -> M=v, lanes 16-31 -> M=v+8; N=col
#pragma unroll
        for (int v = 0; v < 8; ++v) {
            const int m = v + half * 8;
            orow[m * ENC_DIM + c] = acc[v] * scale;
        }
    }
}

extern "C" void kernel_launch(void* const* d_in, const int* in_sizes, int n_in,
                              void* d_out, int out_size, void* d_ws, size_t ws_size,
                              hipStream_t stream) {
    const float* inp       = (const float*)d_in[0];
    const int*   history   = (const int*)  d_in[1];
    const int*   depth     = (const int*)  d_in[2];
    const float* bbox_emb  = (const float*)d_in[3];
    const float* nodes_min = (const float*)d_in[4];
    const float* nodes_max = (const float*)d_in[5];
    float*       out       = (float*)d_out;

    const int n_rays = in_sizes[2];                   // depth has one entry per ray
    int* d_maxd = (int*)d_ws;

    hipMemsetAsync(d_maxd, 0, sizeof(int), stream);   // capture-safe memset node
    depth_max_kernel<<<(n_rays + 255) / 256, 256, 0, stream>>>(depth, n_rays, d_maxd);
    bbox_encode_kernel<<<n_rays, 256, 0, stream>>>(inp, history, bbox_emb,
                                                   nodes_min, nodes_max,
                                                   d_maxd, out);
}